// MixtureOfExperts_35742717837362
// MI455X (gfx1250) — compile-verified
//
#include <hip/hip_runtime.h>

// ---------------------------------------------------------------------------
// MixtureOfExperts for MI455X (gfx1250).
// D = 256, B = E = H = IN = 256, NUM_HEADS = 4, hd = 64.
//
// K1: per-expert fused MLP: two 256^3 f32 GEMMs via V_WMMA_F32_16X16X4_F32,
//     2Mx2N register blocking; intermediate in stride-258 LDS (8B-aligned
//     rows -> ds_load_b64 A-fragments, bank-conflict-free).
// K2: per-batch-row attention-over-experts. eo[b] staged into stride-257 LDS
//     by ONE Tensor-Data-Mover descriptor (pad_interval=256dw, pad=1dw),
//     fenced with s_wait_tensorcnt. Wk folded through q (kills the
//     8.6-GFLOP k-projection GEMM and the 64 MB k tensor).
// ---------------------------------------------------------------------------

typedef __attribute__((ext_vector_type(2))) float v2f;
typedef __attribute__((ext_vector_type(8))) float v8f;
typedef __attribute__((ext_vector_type(4))) unsigned int u32x4;
typedef __attribute__((ext_vector_type(4))) int i32x4;
typedef __attribute__((ext_vector_type(8))) int i32x8;

#define Dm 256
#define K1_STRIDE 258   // even -> 8B-aligned rows; (2*row+i)%64 bank-clean
#define K2_STRIDE 257   // produced by TDM pad feature (1 DWORD per 256)

__device__ __forceinline__ v8f wmma_f32_k4(v2f a, v2f b, v8f c) {
  return __builtin_amdgcn_wmma_f32_16x16x4_f32(
      false, a, false, b, (short)0, c, false, false);
}

// ===========================================================================
// Kernel 1: expert MLP.  One workgroup (8 waves) per expert.
// Wave w owns M-tiles {2w, 2w+1}; loops over 8 N-tile pairs (2x2 blocking).
// ===========================================================================
__global__ __launch_bounds__(256) void moe_expert_mlp(
    const float* __restrict__ X,    // (256,256)
    const float* __restrict__ W1,   // (E,256,256)
    const float* __restrict__ b1,   // (E,256)
    const float* __restrict__ g1,   // (E,256)
    const float* __restrict__ be1,  // (E,256)
    const float* __restrict__ W2,   // (E,256,256)
    const float* __restrict__ b2,   // (E,256)
    float* __restrict__ eo)         // (B,E,H)
{
  extern __shared__ float lds[];    // 256 * 258 floats
  const int e    = blockIdx.x;
  const int tid  = threadIdx.x;
  const int wave = tid >> 5;
  const int lane = tid & 31;
  const int nl   = lane & 15;          // M (A frag) / N (B frag) index
  const int kb   = (lane >> 4) << 1;   // lanes 0-15: K={0,1}; 16-31: K={2,3}
  const int mhi  = (lane >> 4) << 3;   // C frag: vgpr r -> row r + 8*(lane/16)

  const float* __restrict__ W1e = W1 + (size_t)e * 65536;
  const float* __restrict__ W2e = W2 + (size_t)e * 65536;

  const int b0a = (wave * 2) * 16;       // M-tile 0 row base
  const int b0b = (wave * 2 + 1) * 16;   // M-tile 1 row base

  // ---- GEMM1: T[b,o] = sum_i X[b,i] * W1e[o,i] + b1 -> LDS ----
  {
    const float* ap0 = X + (size_t)(b0a + nl) * Dm + kb;
    const float* ap1 = X + (size_t)(b0b + nl) * Dm + kb;
    for (int nb = 0; nb < 8; ++nb) {
      const int oc0 = nb * 32 + nl;        // N-tile 0 column
      const int oc1 = nb * 32 + 16 + nl;   // N-tile 1 column
      v8f c00 = {}, c01 = {}, c10 = {}, c11 = {};
      const float* bp0 = W1e + (size_t)oc0 * Dm + kb;
      const float* bp1 = W1e + (size_t)oc1 * Dm + kb;
      for (int k0 = 0; k0 < Dm; k0 += 32) {
        __builtin_prefetch(bp0 + k0 + 256, 0, 0);  // stream W1 into GL2
        __builtin_prefetch(bp1 + k0 + 256, 0, 0);
#pragma unroll
        for (int kk = 0; kk < 32; kk += 4) {
          const int k = k0 + kk;
          v2f a0 = *(const v2f*)(ap0 + k);
          v2f a1 = *(const v2f*)(ap1 + k);
          v2f bf0 = *(const v2f*)(bp0 + k);
          v2f bf1 = *(const v2f*)(bp1 + k);
          c00 = wmma_f32_k4(a0, bf0, c00);
          c10 = wmma_f32_k4(a1, bf0, c10);
          c01 = wmma_f32_k4(a0, bf1, c01);
          c11 = wmma_f32_k4(a1, bf1, c11);
        }
      }
      const float bias0 = b1[e * Dm + oc0];
      const float bias1 = b1[e * Dm + oc1];
#pragma unroll
      for (int r = 0; r < 8; ++r) {
        lds[(b0a + mhi + r) * K1_STRIDE + oc0] = c00[r] + bias0;
        lds[(b0a + mhi + r) * K1_STRIDE + oc1] = c01[r] + bias1;
        lds[(b0b + mhi + r) * K1_STRIDE + oc0] = c10[r] + bias0;
        lds[(b0b + mhi + r) * K1_STRIDE + oc1] = c11[r] + bias1;
      }
    }
  }
  __syncthreads();

  // ---- LayerNorm over o (256) + ReLU, one row per thread, v2f walks ----
  {
    float* row = lds + tid * K1_STRIDE;      // 8B-aligned base
    float s = 0.f;
    for (int i = 0; i < Dm; i += 2) {
      v2f rv = *(const v2f*)(row + i);
      s += rv.x + rv.y;
    }
    const float mu = s * (1.f / 256.f);
    float v = 0.f;
    for (int i = 0; i < Dm; i += 2) {
      v2f rv = *(const v2f*)(row + i);
      float d0 = rv.x - mu, d1 = rv.y - mu;
      v += d0 * d0 + d1 * d1;
    }
    const float inv = rsqrtf(v * (1.f / 256.f) + 1e-5f);
    const float* g = g1 + e * Dm;
    const float* b = be1 + e * Dm;
    for (int i = 0; i < Dm; i += 2) {
      v2f rv = *(const v2f*)(row + i);
      v2f gv = *(const v2f*)(g + i);
      v2f bv = *(const v2f*)(b + i);
      v2f yv;
      yv.x = (rv.x - mu) * inv * gv.x + bv.x;
      yv.y = (rv.y - mu) * inv * gv.y + bv.y;
      yv.x = yv.x > 0.f ? yv.x : 0.f;
      yv.y = yv.y > 0.f ? yv.y : 0.f;
      *(v2f*)(row + i) = yv;
    }
  }
  __syncthreads();

  // ---- GEMM2: eo[b,e,o] = sum_h h[b,h] * W2e[o,h] + b2 -> global ----
  {
    const float* la0 = lds + (b0a + nl) * K1_STRIDE + kb;  // 8B-aligned
    const float* la1 = lds + (b0b + nl) * K1_STRIDE + kb;
    for (int nb = 0; nb < 8; ++nb) {
      const int oc0 = nb * 32 + nl;
      const int oc1 = nb * 32 + 16 + nl;
      v8f c00 = {}, c01 = {}, c10 = {}, c11 = {};
      const float* bp0 = W2e + (size_t)oc0 * Dm + kb;
      const float* bp1 = W2e + (size_t)oc1 * Dm + kb;
      for (int k0 = 0; k0 < Dm; k0 += 32) {
        __builtin_prefetch(bp0 + k0 + 256, 0, 0);  // stream W2 into GL2
        __builtin_prefetch(bp1 + k0 + 256, 0, 0);
#pragma unroll
        for (int kk = 0; kk < 32; kk += 4) {
          const int k = k0 + kk;
          v2f a0 = *(const v2f*)(la0 + k);   // ds_load_b64
          v2f a1 = *(const v2f*)(la1 + k);
          v2f bf0 = *(const v2f*)(bp0 + k);
          v2f bf1 = *(const v2f*)(bp1 + k);
          c00 = wmma_f32_k4(a0, bf0, c00);
          c10 = wmma_f32_k4(a1, bf0, c10);
          c01 = wmma_f32_k4(a0, bf1, c01);
          c11 = wmma_f32_k4(a1, bf1, c11);
        }
      }
      const float bias0 = b2[e * Dm + oc0];
      const float bias1 = b2[e * Dm + oc1];
#pragma unroll
      for (int r = 0; r < 8; ++r) {
        eo[(size_t)(b0a + mhi + r) * 65536 + e * Dm + oc0] = c00[r] + bias0;
        eo[(size_t)(b0a + mhi + r) * 65536 + e * Dm + oc1] = c01[r] + bias1;
        eo[(size_t)(b0b + mhi + r) * 65536 + e * Dm + oc0] = c10[r] + bias0;
        eo[(size_t)(b0b + mhi + r) * 65536 + e * Dm + oc1] = c11[r] + bias1;
      }
    }
  }
}

// ===========================================================================
// TDM: load one 256x256 f32 tile (row-major, stride 256) into LDS with a
// 1-DWORD pad every 256 DWORDs -> LDS rows land at stride 257.
// ===========================================================================
__device__ __forceinline__ void tdm_load_tile_256x256_pad(
    const float* __restrict__ src, float* ldsDst) {
  const unsigned long long ga = (unsigned long long)(uintptr_t)src;
  const unsigned int lds_addr = (unsigned int)(uintptr_t)ldsDst;  // addr[31:0]

  u32x4 g0;
  g0.x = 1u;                                            // count=1, user D#
  g0.y = lds_addr;                                      // lds_addr [63:32]
  g0.z = (unsigned int)(ga & 0xFFFFFFFFull);            // global_addr lo
  g0.w = (unsigned int)((ga >> 32) & 0x01FFFFFFull)     // global_addr [56:32]
         | (2u << 30);                                  // type = 2 ("image")

  i32x8 g1;
  g1[0] = (2 << 16)      // data_size = 4 bytes
        | (1 << 20)      // pad_enable
        | (7 << 22)      // pad_interval: 256 DWORDs
        | (0 << 25);     // pad_amount: 1 DWORD
  g1[1] = (int)(256u << 16);   // tensor_dim0[15:0] in bits [63:48]
  g1[2] = (int)(256u << 16);   // tensor_dim1[15:0] in [95:80]
  g1[3] = (int)(256u << 16);   // tile_dim0 = 256 in [127:112]
  g1[4] = 256;                 // tile_dim1 = 256 in [143:128]; tile_dim2=0
  g1[5] = 256;                 // tensor_dim0_stride = 256 (data_size units)
  g1[6] = 0;
  g1[7] = 0;

  const i32x4 z4 = {0, 0, 0, 0};
#if __clang_major__ >= 23
  const i32x8 z8 = {0, 0, 0, 0, 0, 0, 0, 0};
  __builtin_amdgcn_tensor_load_to_lds(g0, g1, z4, z4, z8, 0);
#else
  __builtin_amdgcn_tensor_load_to_lds(g0, g1, z4, z4, 0);
#endif
}

// ===========================================================================
// Kernel 2: attention over experts, one workgroup per batch row b.
// ===========================================================================

__device__ __forceinline__ float block_sum(float v, float* red) {
#pragma unroll
  for (int off = 16; off; off >>= 1) v += __shfl_down(v, off, 32);
  __syncthreads();
  if ((threadIdx.x & 31) == 0) red[threadIdx.x >> 5] = v;
  __syncthreads();
  float t = 0.f;
#pragma unroll
  for (int i = 0; i < 8; ++i) t += red[i];
  return t;
}

__device__ __forceinline__ float block_max(float v, float* red) {
#pragma unroll
  for (int off = 16; off; off >>= 1) v = fmaxf(v, __shfl_down(v, off, 32));
  __syncthreads();
  if ((threadIdx.x & 31) == 0) red[threadIdx.x >> 5] = v;
  __syncthreads();
  float t = -3.402823466e+38f;
#pragma unroll
  for (int i = 0; i < 8; ++i) t = fmaxf(t, red[i]);
  return t;
}

__global__ __launch_bounds__(256) void moe_attend_combine(
    const float* __restrict__ te,    // (B,256)
    const float* __restrict__ inw,   // (768,256): Wq rows 0..255, Wk rows 256..511
    const float* __restrict__ inb,   // (768,)
    const float* __restrict__ cw,    // (256,256)
    const float* __restrict__ cb,
    const float* __restrict__ cg,
    const float* __restrict__ cbe,
    const float* __restrict__ eo,    // (B,E,H)
    float* __restrict__ out,         // (B,256)
    float* __restrict__ wout)        // (B,1,256)
{
  extern __shared__ float s[];
  float* eoS   = s;                       // 256*257
  float* qS    = eoS + 256 * K2_STRIDE;   // 256
  float* teS   = qS + 256;                // 256
  float* pS    = teS + 256;               // 4*256
  float* wS    = pS + 1024;               // 256
  float* combS = wS + 256;                // 256
  float* redS  = combS + 256;             // 32
  float* s0S   = redS + 32;               // 4

  const int b   = blockIdx.x;
  const int tid = threadIdx.x;

  // Stage eo[b] (256x256) into stride-257 LDS with a single TDM descriptor.
  teS[tid] = te[b * Dm + tid];
  if (tid == 0) {
    tdm_load_tile_256x256_pad(eo + (size_t)b * 65536, eoS);
    __builtin_amdgcn_s_wait_tensorcnt(0);
  }
  __syncthreads();

  // q[o] = (te[b]·Wq[o,:] + bq[o]) / sqrt(64)
  {
    float acc = 0.f;
    const float* wr = inw + (size_t)tid * Dm;
    for (int i = 0; i < Dm; i += 4) {
      acc += teS[i] * wr[i] + teS[i + 1] * wr[i + 1] +
             teS[i + 2] * wr[i + 2] + teS[i + 3] * wr[i + 3];
    }
    qS[tid] = (acc + inb[tid]) * 0.125f;
  }
  __syncthreads();

  // s0[n] = q[n,:]·bk[n,:]
  if (tid < 4) {
    float acc = 0.f;
    for (int d = 0; d < 64; ++d)
      acc += qS[tid * 64 + d] * inb[256 + tid * 64 + d];
    s0S[tid] = acc;
  }

  // p[n,h] = sum_d q[n,d] * Wk[n*64+d, h]   (coalesced over h = tid)
  for (int n = 0; n < 4; ++n) {
    float acc = 0.f;
    const float* wk = inw + (size_t)(256 + n * 64) * Dm + tid;
    for (int d = 0; d < 64; ++d) acc += qS[n * 64 + d] * wk[(size_t)d * Dm];
    pS[n * Dm + tid] = acc;
  }
  __syncthreads();

  // scores[n, e=tid] = eo[b,e,:]·p[n,:] + s0[n]
  float sc0 = s0S[0], sc1 = s0S[1], sc2 = s0S[2], sc3 = s0S[3];
  {
    const float* er = eoS + tid * K2_STRIDE;
    for (int h = 0; h < Dm; ++h) {
      const float ev = er[h];
      sc0 += ev * pS[h];
      sc1 += ev * pS[256 + h];
      sc2 += ev * pS[512 + h];
      sc3 += ev * pS[768 + h];
    }
  }

  // softmax over e per head (block-wide), then mean over heads
  float w = 0.f;
  {
    float m0 = block_max(sc0, redS); float e0 = __expf(sc0 - m0);
    float z0 = block_sum(e0, redS);  w += e0 / z0;
    float m1 = block_max(sc1, redS); float e1 = __expf(sc1 - m1);
    float z1 = block_sum(e1, redS);  w += e1 / z1;
    float m2 = block_max(sc2, redS); float e2 = __expf(sc2 - m2);
    float z2 = block_sum(e2, redS);  w += e2 / z2;
    float m3 = block_max(sc3, redS); float e3 = __expf(sc3 - m3);
    float z3 = block_sum(e3, redS);  w += e3 / z3;
    w *= 0.25f;
  }
  wS[tid] = w;
  wout[(size_t)b * Dm + tid] = w;
  __syncthreads();

  // combined[h=tid] = sum_e eo[b,e,h] * w[e]   (stride-257 -> bank-clean)
  {
    float acc = 0.f;
    for (int e2 = 0; e2 < Dm; ++e2)
      acc += eoS[e2 * K2_STRIDE + tid] * wS[e2];
    combS[tid] = acc;
  }
  __syncthreads();

  // out = relu(LN(combined @ cw^T + cb) * cg + cbeta)
  {
    float y = cb[tid];
    const float* wr = cw + (size_t)tid * Dm;
    for (int i = 0; i < Dm; i += 4) {
      y += combS[i] * wr[i] + combS[i + 1] * wr[i + 1] +
           combS[i + 2] * wr[i + 2] + combS[i + 3] * wr[i + 3];
    }
    const float mu = block_sum(y, redS) * (1.f / 256.f);
    const float d  = y - mu;
    const float var = block_sum(d * d, redS) * (1.f / 256.f);
    float o = d * rsqrtf(var + 1e-5f) * cg[tid] + cbe[tid];
    out[(size_t)b * Dm + tid] = o > 0.f ? o : 0.f;
  }
}

// ===========================================================================
extern "C" void kernel_launch(void* const* d_in, const int* in_sizes, int n_in,
                              void* d_out, int out_size, void* d_ws, size_t ws_size,
                              hipStream_t stream) {
  const float* x   = (const float*)d_in[0];
  const float* te  = (const float*)d_in[1];
  const float* W1  = (const float*)d_in[2];
  const float* b1  = (const float*)d_in[3];
  const float* g1  = (const float*)d_in[4];
  const float* be1 = (const float*)d_in[5];
  const float* W2  = (const float*)d_in[6];
  const float* b2  = (const float*)d_in[7];
  const float* inw = (const float*)d_in[8];
  const float* inb = (const float*)d_in[9];
  const float* cw  = (const float*)d_in[10];
  const float* cb  = (const float*)d_in[11];
  const float* cg  = (const float*)d_in[12];
  const float* cbe = (const float*)d_in[13];

  float* out  = (float*)d_out;        // (256,256)
  float* wout = out + 65536;          // (256,1,256)
  float* eo   = out + 2 * 65536;      // (256,256,256)

  const size_t lds1 = (size_t)256 * K1_STRIDE * sizeof(float);
  moe_expert_mlp<<<256, 256, lds1, stream>>>(x, W1, b1, g1, be1, W2, b2, eo);

  const size_t lds2 =
      ((size_t)256 * K2_STRIDE + 256 + 256 + 1024 + 256 + 256 + 32 + 4) *
      sizeof(float);
  moe_attend_combine<<<256, 256, lds2, stream>>>(te, inw, inb, cw, cb, cg, cbe,
                                                 eo, out, wout);
}